// DetectionEvalWrapper_42975442764577
// MI455X (gfx1250) — compile-verified
//
#include <hip/hip_runtime.h>
#include <hip/hip_bf16.h>
#include <stdint.h>

// ---------------- problem constants (from reference) ----------------
#define NIMG        8
#define NANCH       49104
#define NCLS        90
#define NELEMS      (NANCH * NCLS)        // 4,419,360 logits per image
#define NDETPTS     5000                  // MAX_DET_POINTS
#define NDET        100                   // MAX_DETECTIONS
#define NBINS       2048                  // 11-bit radix level
#define TILE_ELEMS  4096                  // 16 KB TDM tile
#define TILE_BYTES  (TILE_ELEMS * 4)
#define NTILES      ((NELEMS + TILE_ELEMS - 1) / TILE_ELEMS)   // 1079
#define SPILL_CAP   16384
#define HIST_LDS_OFF (2 * TILE_BYTES)     // hist after the two tile buffers

// ---------------- CDNA5 feature detection ----------------
#if defined(__has_builtin)
#  if __has_builtin(__builtin_amdgcn_tensor_load_to_lds) && __has_builtin(__builtin_amdgcn_s_wait_tensorcnt)
#    define HAVE_TDM 1
#  else
#    define HAVE_TDM 0
#  endif
#else
#  define HAVE_TDM 0
#endif

#if HAVE_TDM
#pragma message("CDNA5: TDM tensor_load_to_lds path ENABLED in this pass")
#else
#pragma message("CDNA5: TDM path disabled in this pass (host pass or builtin missing)")
#endif

typedef unsigned int v4u __attribute__((ext_vector_type(4)));
typedef int          v8i __attribute__((ext_vector_type(8)));
typedef int          v4i __attribute__((ext_vector_type(4)));

// float -> order-preserving unsigned key (bigger float => bigger key)
__device__ __forceinline__ unsigned orderKey(float f) {
    unsigned u = __float_as_uint(f);
    return (u & 0x80000000u) ? ~u : (u | 0x80000000u);
}

#if HAVE_TDM
// Issue a 1-D TDM load of nElems f32 from gptr into LDS offset ldsOff.
// D# per CDNA5 ISA §8.3/8.4: group0 = {count/flags, lds_addr, global_addr, type=2},
// group1 = {data_size=4B, tensor_dim0=n, tensor_dim1=1, tile_dim0=n, tile_dim1=1, stride0=n}.
__device__ __forceinline__ void tdm_load_f32_1d(unsigned ldsOff, const void* gptr, unsigned nElems) {
    unsigned long long ga = (unsigned long long)gptr;
    v4u g0;
    g0[0] = 1u;                                                  // count=1, user mode, no gather
    g0[1] = ldsOff;                                              // lds_addr (bytes)
    g0[2] = (unsigned)(ga & 0xFFFFFFFFull);                      // global_addr[31:0]
    g0[3] = (unsigned)((ga >> 32) & 0x01FFFFFFull) | (2u << 30); // addr[56:32] | type=2
    v8i g1;
    g1[0] = (int)(2u << 16);                                     // workgroup_mask=0, data_size=2 (4B)
    g1[1] = (int)((nElems & 0xFFFFu) << 16);                     // tensor_dim0[15:0]
    g1[2] = (int)(1u << 16);                                     // tensor_dim0 hi | tensor_dim1 lo=1
    g1[3] = (int)((nElems & 0xFFFFu) << 16);                     // tensor_dim1 hi | tile_dim0=n
    g1[4] = 1;                                                   // tile_dim1=1, tile_dim2=0
    g1[5] = (int)nElems;                                         // tensor_dim0_stride lo32
    g1[6] = 0;
    g1[7] = 0;
    v4i z = {0, 0, 0, 0};
#if __clang_major__ >= 23
    v8i z8 = {0, 0, 0, 0, 0, 0, 0, 0};
    __builtin_amdgcn_tensor_load_to_lds(g0, g1, z, z, z8, 0);
#else
    __builtin_amdgcn_tensor_load_to_lds(g0, g1, z, z, 0);
#endif
}
#endif

// ---------------- K1: level-1 histogram (double-buffered TDM stream) ----------------
// dynamic LDS layout: [0,16K) tileA | [16K,32K) tileB | [32K,40K) 2048-bin hist
__global__ void k_hist1(const float* __restrict__ cls, unsigned* __restrict__ hist1) {
    extern __shared__ char smem[];
    unsigned* lh = (unsigned*)(smem + HIST_LDS_OFF);
    const int b = blockIdx.y;
    const float* base = cls + (size_t)b * NELEMS;

    for (int i = threadIdx.x; i < NBINS; i += blockDim.x) lh[i] = 0u;
    __syncthreads();

#if HAVE_TDM
    const int stride = gridDim.x;
    const int t0 = blockIdx.x;
    if (t0 < NTILES && threadIdx.x < 32) {                 // prime the pipeline (wave 0)
        const int n0 = min(TILE_ELEMS, NELEMS - t0 * TILE_ELEMS);
        tdm_load_f32_1d(0u, base + (size_t)t0 * TILE_ELEMS, (unsigned)n0);
    }
    int parity = 0;
    for (int t = t0; t < NTILES; t += stride, parity ^= 1) {
        const int n  = min(TILE_ELEMS, NELEMS - t * TILE_ELEMS);
        const int tn = t + stride;
        if (threadIdx.x < 32) {
            if (tn < NTILES) {                             // overlap next DMA with this consume
                const int nn = min(TILE_ELEMS, NELEMS - tn * TILE_ELEMS);
                tdm_load_f32_1d((unsigned)((parity ^ 1) * TILE_BYTES),
                                base + (size_t)tn * TILE_ELEMS, (unsigned)nn);
                __builtin_amdgcn_s_wait_tensorcnt(1);      // tile t landed (in-order)
            } else {
                __builtin_amdgcn_s_wait_tensorcnt(0);
            }
        }
        __syncthreads();
        const float4* tp = (const float4*)(smem + parity * TILE_BYTES);
        for (int i = threadIdx.x; i < (n >> 2); i += blockDim.x) {
            float4 v = tp[i];                              // ds_load_b128
            atomicAdd(&lh[orderKey(v.x) >> 21], 1u);
            atomicAdd(&lh[orderKey(v.y) >> 21], 1u);
            atomicAdd(&lh[orderKey(v.z) >> 21], 1u);
            atomicAdd(&lh[orderKey(v.w) >> 21], 1u);
        }
        __syncthreads();                                   // buffer safe to overwrite next iter
    }
#else
    const float4* vbase = (const float4*)base;
    const int nv = NELEMS / 4;
    for (int j = blockIdx.x * blockDim.x + threadIdx.x; j < nv; j += gridDim.x * blockDim.x) {
        float4 v = vbase[j];
        atomicAdd(&lh[orderKey(v.x) >> 21], 1u);
        atomicAdd(&lh[orderKey(v.y) >> 21], 1u);
        atomicAdd(&lh[orderKey(v.z) >> 21], 1u);
        atomicAdd(&lh[orderKey(v.w) >> 21], 1u);
    }
#endif
    __syncthreads();
    unsigned* gh = hist1 + (size_t)b * NBINS;
    for (int i = threadIdx.x; i < NBINS; i += blockDim.x) {
        unsigned v = lh[i];
        if (v) atomicAdd(&gh[i], v);
    }
}

// ---------------- K2: scan level-1 ----------------
__global__ void k_scanA(const unsigned* __restrict__ hist1, int2* __restrict__ params1) {
    const int b = blockIdx.x;
    if (threadIdx.x != 0) return;
    const unsigned* h = hist1 + (size_t)b * NBINS;
    unsigned cum = 0, above = 0;
    int T1 = 0;
    for (int i = NBINS - 1; i >= 0; --i) {
        unsigned c = h[i];
        if (cum + c >= (unsigned)NDETPTS) { T1 = i; above = cum; break; }
        cum += c; above = cum;
    }
    params1[b] = make_int2(T1, NDETPTS - (int)above);
}

// ---------------- K3: compact > T1, spill == T1, level-2 histogram ----------------
__global__ void k_passB(const float* __restrict__ cls, const int2* __restrict__ params1,
                        float* __restrict__ candVal, unsigned* __restrict__ candIdx,
                        unsigned* __restrict__ candCount,
                        float* __restrict__ spillVal, unsigned* __restrict__ spillIdx,
                        unsigned* __restrict__ spillCount, unsigned* __restrict__ hist2) {
    const int b  = blockIdx.y;
    const int T1 = params1[b].x;
    const float4* base = (const float4*)(cls + (size_t)b * NELEMS);
    unsigned* h2 = hist2 + (size_t)b * NBINS;
    const int nv = NELEMS / 4;
    for (int j = blockIdx.x * blockDim.x + threadIdx.x; j < nv; j += gridDim.x * blockDim.x) {
        float4 v = base[j];                                // L2-resident after pass A
        float vals[4] = {v.x, v.y, v.z, v.w};
#pragma unroll
        for (int k = 0; k < 4; ++k) {
            unsigned key = orderKey(vals[k]);
            int bin = (int)(key >> 21);
            if (bin > T1) {
                unsigned p = atomicAdd(&candCount[b], 1u);
                if (p < (unsigned)NDETPTS) {
                    candVal[(size_t)b * NDETPTS + p] = vals[k];
                    candIdx[(size_t)b * NDETPTS + p] = (unsigned)(4 * j + k);
                }
            } else if (bin == T1) {
                atomicAdd(&h2[(key >> 10) & 0x7FFu], 1u);
                unsigned p = atomicAdd(&spillCount[b], 1u);
                if (p < (unsigned)SPILL_CAP) {
                    spillVal[(size_t)b * SPILL_CAP + p] = vals[k];
                    spillIdx[(size_t)b * SPILL_CAP + p] = (unsigned)(4 * j + k);
                }
            }
        }
    }
}

// ---------------- K4: scan level-2, build level-3 from spill, exact K* ----------------
__global__ void k_scanB(const unsigned* __restrict__ hist2, const int2* __restrict__ params1,
                        const float* __restrict__ spillVal, const unsigned* __restrict__ spillCount,
                        uint2* __restrict__ params2) {
    __shared__ unsigned h3[1024];
    __shared__ int sT2, sNeed2;
    const int b = blockIdx.x;
    const int T1 = params1[b].x, need1 = params1[b].y;
    if (threadIdx.x == 0) {
        const unsigned* h = hist2 + (size_t)b * NBINS;
        unsigned cum = 0, above = 0;
        int T2 = 0;
        for (int i = NBINS - 1; i >= 0; --i) {
            unsigned c = h[i];
            if (cum + c >= (unsigned)need1) { T2 = i; above = cum; break; }
            cum += c; above = cum;
        }
        sT2 = T2; sNeed2 = need1 - (int)above;
    }
    for (int i = threadIdx.x; i < 1024; i += blockDim.x) h3[i] = 0u;
    __syncthreads();
    const int T2 = sT2, need2 = sNeed2;
    const unsigned cnt = min(spillCount[b], (unsigned)SPILL_CAP);
    const float* sv = spillVal + (size_t)b * SPILL_CAP;
    for (unsigned i = threadIdx.x; i < cnt; i += blockDim.x) {
        unsigned k = orderKey(sv[i]);
        if ((int)((k >> 10) & 0x7FFu) == T2) atomicAdd(&h3[k & 0x3FFu], 1u);
    }
    __syncthreads();
    if (threadIdx.x == 0) {
        unsigned cum = 0, above = 0;
        int T3 = 0;
        for (int i = 1023; i >= 0; --i) {
            unsigned c = h3[i];
            if (cum + c >= (unsigned)need2) { T3 = i; above = cum; break; }
            cum += c; above = cum;
        }
        unsigned Kstar = ((unsigned)T1 << 21) | ((unsigned)T2 << 10) | (unsigned)T3;
        params2[b] = make_uint2(Kstar, (unsigned)(need2 - (int)above));
    }
}

// ---------------- K5: finish compaction from spill ----------------
__global__ void k_passC(const float* __restrict__ spillVal, const unsigned* __restrict__ spillIdx,
                        const unsigned* __restrict__ spillCount, const uint2* __restrict__ params2,
                        float* __restrict__ candVal, unsigned* __restrict__ candIdx,
                        unsigned* __restrict__ candCount, unsigned* __restrict__ eqCount) {
    const int b = blockIdx.x;
    const unsigned Kstar = params2[b].x, needEq = params2[b].y;
    const unsigned cnt = min(spillCount[b], (unsigned)SPILL_CAP);
    const float* sv = spillVal + (size_t)b * SPILL_CAP;
    const unsigned* si = spillIdx + (size_t)b * SPILL_CAP;
    for (unsigned i = threadIdx.x; i < cnt; i += blockDim.x) {
        float v = sv[i];
        unsigned k = orderKey(v);
        bool take = (k > Kstar);
        if (!take && k == Kstar) take = (atomicAdd(&eqCount[b], 1u) < needEq);
        if (take) {
            unsigned p = atomicAdd(&candCount[b], 1u);
            if (p < (unsigned)NDETPTS) {
                candVal[(size_t)b * NDETPTS + p] = v;
                candIdx[(size_t)b * NDETPTS + p] = si[i];
            }
        }
    }
}

// ---------------- K6: decode + sigmoid ----------------
__global__ void k_decode(const float* __restrict__ box_out, const float* __restrict__ anchors,
                         const float* __restrict__ candVal, const unsigned* __restrict__ candIdx,
                         float4* __restrict__ boxes, float* __restrict__ scores,
                         int* __restrict__ clsArr) {
    const int b = blockIdx.y;
    const int i = blockIdx.x * blockDim.x + threadIdx.x;
    if (i >= NDETPTS) return;
    const size_t o = (size_t)b * NDETPTS + i;
    const float logit = candVal[o];
    const unsigned flat = candIdx[o];
    const unsigned a = flat / (unsigned)NCLS;
    const int c = (int)(flat - a * (unsigned)NCLS);
    float4 an = ((const float4*)anchors)[a];                     // (ymin,xmin,ymax,xmax)
    float4 rb = ((const float4*)box_out)[(size_t)b * NANCH + a]; // (ty,tx,th,tw)
    float ycA = (an.x + an.z) * 0.5f, xcA = (an.y + an.w) * 0.5f;
    float ha = an.z - an.x, wa = an.w - an.y;
    float h = expf(rb.z) * ha, w = expf(rb.w) * wa;
    float yc = rb.x * ha + ycA, xc = rb.y * wa + xcA;
    boxes[o]  = make_float4(yc - h * 0.5f, xc - w * 0.5f, yc + h * 0.5f, xc + w * 0.5f);
    scores[o] = 1.0f / (1.0f + expf(-logit));
    clsArr[o] = c;
}

// ---------------- K7: greedy NMS (one 1024-thread block per image) ----------------
__global__ void k_nms(const float4* __restrict__ boxes, const float* __restrict__ scores,
                      const int* __restrict__ clsArr, int* __restrict__ keep) {
    __shared__ float sc[NDETPTS];
    __shared__ unsigned char scls[NDETPTS];   // class in [0,90): byte is enough
    __shared__ float wv[32];
    __shared__ int   wi[32];
    __shared__ float4 s_box;
    __shared__ int    s_cls;
    const int b = blockIdx.x;
    const float4* bx = boxes + (size_t)b * NDETPTS;
    for (int i = threadIdx.x; i < NDETPTS; i += blockDim.x) {
        sc[i]   = scores[(size_t)b * NDETPTS + i];
        scls[i] = (unsigned char)clsArr[(size_t)b * NDETPTS + i];
    }
    __syncthreads();

    for (int it = 0; it < NDET; ++it) {
        float bv = -2.0f; int bi = 0x7fffffff;
        for (int i = threadIdx.x; i < NDETPTS; i += blockDim.x) {
            float v = sc[i];
            if (v > bv || (v == bv && i < bi)) { bv = v; bi = i; }
        }
        for (int off = 16; off > 0; off >>= 1) {             // wave32 reduce
            float ov = __shfl_down(bv, off, 32);
            int   oi = __shfl_down(bi, off, 32);
            if (ov > bv || (ov == bv && oi < bi)) { bv = ov; bi = oi; }
        }
        const int lane = threadIdx.x & 31, wid = threadIdx.x >> 5;
        if (lane == 0) { wv[wid] = bv; wi[wid] = bi; }
        __syncthreads();
        if (threadIdx.x < 32) {
            bv = wv[threadIdx.x]; bi = wi[threadIdx.x];
            for (int off = 16; off > 0; off >>= 1) {
                float ov = __shfl_down(bv, off, 32);
                int   oi = __shfl_down(bi, off, 32);
                if (ov > bv || (ov == bv && oi < bi)) { bv = ov; bi = oi; }
            }
            if (threadIdx.x == 0) {
                keep[(size_t)b * NDET + it] = bi;
                s_box = bx[bi];
                s_cls = (int)scls[bi];
            }
        }
        __syncthreads();
        const float4 kb = s_box;
        const int    kc = s_cls;
        const float  ka = (kb.z - kb.x) * (kb.w - kb.y);
        for (int i = threadIdx.x; i < NDETPTS; i += blockDim.x) {
            if (sc[i] <= -1.0f) continue;
            if ((int)scls[i] != kc) continue;   // CLASS_OFFSET trick: cross-class IoU == 0
            float4 q = bx[i];
            float yi0 = fmaxf(kb.x, q.x), xi0 = fmaxf(kb.y, q.y);
            float yi1 = fminf(kb.z, q.z), xi1 = fminf(kb.w, q.w);
            float inter = fmaxf(yi1 - yi0, 0.0f) * fmaxf(xi1 - xi0, 0.0f);
            float a2 = (q.z - q.x) * (q.w - q.y);
            float iou = inter / (ka + a2 - inter + 1e-8f);
            if (iou > 0.5f) sc[i] = -1.0f;      // includes the picked box itself (IoU=1)
        }
        __syncthreads();
    }
}

// ---------------- K8: gather detections ----------------
__global__ void k_out(const float4* __restrict__ boxes, const float* __restrict__ scores,
                      const int* __restrict__ clsArr, const int* __restrict__ keep,
                      const float* __restrict__ scales, float* __restrict__ out) {
    const int b = blockIdx.x, j = threadIdx.x;
    if (j >= NDET) return;
    const int i = keep[(size_t)b * NDET + j];
    const float4 bo = boxes[(size_t)b * NDETPTS + i];
    const float s = scales[b];
    float* d = out + ((size_t)b * NDET + j) * 6;
    d[0] = bo.y * s;             // xmin
    d[1] = bo.x * s;             // ymin
    d[2] = (bo.w - bo.y) * s;    // width
    d[3] = (bo.z - bo.x) * s;    // height
    d[4] = scores[(size_t)b * NDETPTS + i];
    d[5] = (float)clsArr[(size_t)b * NDETPTS + i] + 1.0f;
}

// ---------------- host side ----------------
extern "C" void kernel_launch(void* const* d_in, const int* in_sizes, int n_in,
                              void* d_out, int out_size, void* d_ws, size_t ws_size,
                              hipStream_t stream) {
    (void)in_sizes; (void)n_in; (void)out_size; (void)ws_size;
    const float* cls  = (const float*)d_in[0];   // [8, 49104, 90]
    const float* box  = (const float*)d_in[1];   // [8, 49104, 4]
    const float* anch = (const float*)d_in[2];   // [49104, 4]
    const float* scal = (const float*)d_in[3];   // [8]

    char* ws = (char*)d_ws;
    size_t off = 0;
    auto alloc = [&](size_t sz) { size_t o = off; off = (off + sz + 255) & ~(size_t)255; return o; };
    const size_t oH1  = alloc((size_t)NIMG * NBINS * 4);
    const size_t oH2  = alloc((size_t)NIMG * NBINS * 4);
    const size_t oCnt = alloc((size_t)3 * NIMG * 4);         // candCount | spillCount | eqCount
    const size_t zeroBytes = off;                            // region memset to 0 each call
    const size_t oP1  = alloc((size_t)NIMG * 8);
    const size_t oP2  = alloc((size_t)NIMG * 8);
    const size_t oCV  = alloc((size_t)NIMG * NDETPTS * 4);
    const size_t oCI  = alloc((size_t)NIMG * NDETPTS * 4);
    const size_t oBX  = alloc((size_t)NIMG * NDETPTS * 16);
    const size_t oSC  = alloc((size_t)NIMG * NDETPTS * 4);
    const size_t oCL  = alloc((size_t)NIMG * NDETPTS * 4);
    const size_t oKP  = alloc((size_t)NIMG * NDET * 4);
    const size_t oSV  = alloc((size_t)NIMG * SPILL_CAP * 4);
    const size_t oSI  = alloc((size_t)NIMG * SPILL_CAP * 4);

    unsigned* hist1      = (unsigned*)(ws + oH1);
    unsigned* hist2      = (unsigned*)(ws + oH2);
    unsigned* candCount  = (unsigned*)(ws + oCnt);
    unsigned* spillCount = candCount + NIMG;
    unsigned* eqCount    = candCount + 2 * NIMG;
    int2*     params1    = (int2*)(ws + oP1);
    uint2*    params2    = (uint2*)(ws + oP2);
    float*    candVal    = (float*)(ws + oCV);
    unsigned* candIdx    = (unsigned*)(ws + oCI);
    float4*   boxesB     = (float4*)(ws + oBX);
    float*    scores     = (float*)(ws + oSC);
    int*      clsArr     = (int*)(ws + oCL);
    int*      keep       = (int*)(ws + oKP);
    float*    spillVal   = (float*)(ws + oSV);
    unsigned* spillIdx   = (unsigned*)(ws + oSI);

    hipMemsetAsync(ws, 0, zeroBytes, stream);

    // 40 KB dynamic LDS: two 16 KB TDM tiles + 8 KB histogram
    k_hist1 <<<dim3(64, NIMG), 256, HIST_LDS_OFF + NBINS * 4, stream>>>(cls, hist1);
    k_scanA <<<NIMG, 32, 0, stream>>>(hist1, params1);
    k_passB <<<dim3(512, NIMG), 256, 0, stream>>>(cls, params1, candVal, candIdx, candCount,
                                                  spillVal, spillIdx, spillCount, hist2);
    k_scanB <<<NIMG, 256, 0, stream>>>(hist2, params1, spillVal, spillCount, params2);
    k_passC <<<NIMG, 256, 0, stream>>>(spillVal, spillIdx, spillCount, params2,
                                       candVal, candIdx, candCount, eqCount);
    k_decode<<<dim3((NDETPTS + 255) / 256, NIMG), 256, 0, stream>>>(box, anch, candVal, candIdx,
                                                                    boxesB, scores, clsArr);
    k_nms   <<<NIMG, 1024, 0, stream>>>(boxesB, scores, clsArr, keep);
    k_out   <<<NIMG, 128, 0, stream>>>(boxesB, scores, clsArr, keep, scal, (float*)d_out);
}